// test_22153441312926
// MI455X (gfx1250) — compile-verified
//
#include <hip/hip_runtime.h>
#include <math.h>

// ---------------------------------------------------------------------------
// Edge-filter GNN pipeline for MI455X (gfx1250, wave32).
//
// Stage 1 (node_kernel): x1 = sigmoid(x @ W1^T + b1) via V_WMMA_F32_16X16X4_F32
//   (K=3 padded to 4, 10 output channels padded to 16; b1 preloaded into the
//   C accumulator). One 16x16 tile per wave, spilled through LDS; then one
//   thread per node computes the derived per-node scalars:
//     q  = ||x1||^2
//     s1 = dot(Wx[0:10],  x1)   (contribution when node is an edge source)
//     s2 = dot(Wx[10:20], x1)   (contribution when node is an edge dest)
//   Record layout (16 floats, 64B aligned): [x1[0..9], q, s1, s2, pad, pad, pad]
//
// Stage 2 (edge_kernel): 1 thread/edge. Gather two 64B records (table is
//   ~8MB -> fully resident in the 192MB L2, so gathers are L2-bandwidth).
//     ||h_s - h_d||^2 = q_s + q_d - 2 * dot(h_s, h_d)
//     out = (n2 < 0.25) ? (s1_s + s2_d + bx) : 0
//   HBM traffic ~ 64MB indices + 16MB out -> ~3.5us at 23.3 TB/s.
// ---------------------------------------------------------------------------

typedef __attribute__((ext_vector_type(2))) float v2f;
typedef __attribute__((ext_vector_type(8))) float v8f;

__global__ __launch_bounds__(256) void node_kernel(
    const float* __restrict__ x,    // [N,3]
    const float* __restrict__ W1,   // [10,3] row-major
    const float* __restrict__ b1,   // [10]
    const float* __restrict__ Wx,   // [20]
    float* __restrict__ table,      // [N,16] scratch records
    int N)
{
    __shared__ float lds[8][16][16];   // per-wave 16x16 f32 tile (8 KB)

    const int tid  = threadIdx.x;
    const int wave = tid >> 5;
    const int lane = tid & 31;
    const int l15  = lane & 15;
    const bool hi  = lane >= 16;       // lanes 16-31 carry K=2,3 halves
    const int base = blockIdx.x * 128 + wave * 16;

    // ---- A matrix: 16 rows of x, K = {x0,x1,x2,0}  (ISA 16x4 f32 layout:
    //      lanes 0-15: v0=K0, v1=K1 ; lanes 16-31: v0=K2, v1=K3)
    int m  = base + l15;
    int mc = (m < N) ? m : (N - 1);    // clamp (keeps EXEC all-ones for WMMA)
    v2f a;
    a.x = hi ? x[mc * 3 + 2] : x[mc * 3 + 0];
    a.y = hi ? 0.0f          : x[mc * 3 + 1];

    // ---- B matrix: 4x16 = W1^T zero-padded (B[k][n] = W1[n][k], n<10, k<3)
    //      lanes 0-15: v0=row K0, v1=row K1 ; lanes 16-31: v0=row K2, v1=row K3(=0)
    v2f b;
    b.x = (l15 < 10) ? W1[l15 * 3 + (hi ? 2 : 0)] : 0.0f;
    b.y = (l15 < 10 && !hi) ? W1[l15 * 3 + 1] : 0.0f;

    // ---- C matrix: broadcast bias b1 down every row (column n gets b1[n])
    float bb = (l15 < 10) ? b1[l15] : 0.0f;
    v8f c;
#pragma unroll
    for (int r = 0; r < 8; ++r) c[r] = bb;

    // D = A*B + C  -> pre-activation z for 16 nodes x 16 channels
    v8f dmat = __builtin_amdgcn_wmma_f32_16x16x4_f32(
        /*neg_a=*/false, a, /*neg_b=*/false, b,
        /*c_mod=*/(short)0, c, /*reuse_a=*/false, /*reuse_b=*/false);

    // D VGPR r holds row M = r + (hi?8:0), column N = l15
#pragma unroll
    for (int r = 0; r < 8; ++r)
        lds[wave][r + (hi ? 8 : 0)][l15] = dmat[r];

    __syncthreads();

    // ---- per-node epilogue: sigmoid + derived scalars, 1 thread per node
    if (tid < 128) {
        int node = blockIdx.x * 128 + tid;
        if (node < N) {
            const float* z = lds[tid >> 4][tid & 15];
            float v[10];
            float q = 0.0f, p1 = 0.0f, p2 = 0.0f;
#pragma unroll
            for (int i = 0; i < 10; ++i) {
                float s = 1.0f / (1.0f + expf(-z[i]));
                v[i] = s;
                q  += s * s;
                p1 += Wx[i]      * s;
                p2 += Wx[10 + i] * s;
            }
            float4* row = (float4*)(table + (size_t)node * 16);
            row[0] = make_float4(v[0], v[1], v[2], v[3]);
            row[1] = make_float4(v[4], v[5], v[6], v[7]);
            row[2] = make_float4(v[8], v[9], q, p1);
            row[3] = make_float4(p2, 0.0f, 0.0f, 0.0f);
        }
    }
}

__global__ __launch_bounds__(256) void edge_kernel(
    const long long* __restrict__ ei,  // [2,E] int64
    const float* __restrict__ table,   // [N,16]
    const float* __restrict__ bxp,     // [1]
    float* __restrict__ out,           // [E]
    int E)
{
    int e = blockIdx.x * 256 + threadIdx.x;
    if (e >= E) return;

    // streamed once -> non-temporal loads keep L2 for the node table
    int s = (int)__builtin_nontemporal_load(&ei[e]);
    int d = (int)__builtin_nontemporal_load(&ei[(size_t)E + e]);

    const float4* rs = (const float4*)(table + (size_t)s * 16);
    const float4* rd = (const float4*)(table + (size_t)d * 16);
    float4 a0 = rs[0], a1 = rs[1], a2 = rs[2];
    float4 c0 = rd[0], c1 = rd[1], c2 = rd[2], c3 = rd[3];

    float dot = a0.x * c0.x + a0.y * c0.y + a0.z * c0.z + a0.w * c0.w
              + a1.x * c1.x + a1.y * c1.y + a1.z * c1.z + a1.w * c1.w
              + a2.x * c2.x + a2.y * c2.y;

    float n2  = a2.z + c2.z - 2.0f * dot;   // ||h_s - h_d||^2
    float val = a2.w + c3.x + bxp[0];       // s1_src + s2_dst + bx
    float r   = (n2 < 0.25f) ? val : 0.0f;  // norm < 0.5  <=>  n2 < 0.25

    __builtin_nontemporal_store(r, &out[e]);
}

extern "C" void kernel_launch(void* const* d_in, const int* in_sizes, int n_in,
                              void* d_out, int out_size, void* d_ws, size_t ws_size,
                              hipStream_t stream) {
    // setup_inputs order: x, edge_index, W1, b1, W2, b2, Wx, bx
    const float*     x   = (const float*)d_in[0];
    const long long* ei  = (const long long*)d_in[1];
    const float*     W1  = (const float*)d_in[2];
    const float*     b1  = (const float*)d_in[3];
    const float*     Wx  = (const float*)d_in[6];
    const float*     bx  = (const float*)d_in[7];
    float*           out = (float*)d_out;
    float*           tab = (float*)d_ws;   // needs N*16*4 = 8 MB scratch

    const int N = in_sizes[0] / 3;
    const int E = in_sizes[1] / 2;

    node_kernel<<<(N + 127) / 128, 256, 0, stream>>>(x, W1, b1, Wx, tab, N);
    edge_kernel<<<(E + 255) / 256, 256, 0, stream>>>(ei, tab, bx, out, E);
}